// MeanLoss_59777354826199
// MI455X (gfx1250) — compile-verified
//
#include <hip/hip_runtime.h>

typedef __attribute__((ext_vector_type(2))) float v2f;
typedef __attribute__((ext_vector_type(8))) float v8f;

#define NUM_SEG 64
#define NCH 16
#define NPIX (32 * 512 * 512)      // 8,388,608
#define WGS 256
#define NWG 2048
#define PIX_PER_WG (NPIX / NWG)    // 4096
#define TILE 256
#define TILES_PER_WG (PIX_PER_WG / TILE) // 16
#define LDS_STRIDE 17              // pad to avoid bank conflicts

#define ALPHA 1.0f
#define BETA 1.0f
#define GAMMA 0.001f
#define DELTA_V 0.0f
#define DELTA_D 1.5f

// ---------------------------------------------------------------- zero ws
__global__ void zero_kernel(float* ws, int n, float* out) {
  int i = blockIdx.x * blockDim.x + threadIdx.x;
  if (i < n) ws[i] = 0.0f;
  if (i == 0) out[0] = 0.0f;
}

// ------------------------------------------- pass 1: segment sums via WMMA
// sums[L,C] = OneHot[L,P] x X[P,C] done as f32 16x16x4 WMMAs, 4 label-blocks.
__global__ __launch_bounds__(WGS) void seg_sum_wmma(
    const float* __restrict__ embd, const int* __restrict__ trgt,
    const int* __restrict__ mask, float* __restrict__ sums,
    float* __restrict__ counts) {
  __shared__ float xs[TILE * LDS_STRIDE];          // pixel-major tile [p][c]
  __shared__ __align__(16) int segl[TILE];
  __shared__ int cnt[NUM_SEG];
  __shared__ float red[NUM_SEG * NCH];

  const int t = threadIdx.x;
  const int wave = t >> 5;
  const int lane = t & 31;
  const int m = lane & 15;          // row-in-block for A, channel for B
  const bool hi = lane >= 16;

  for (int i = t; i < NUM_SEG; i += WGS) cnt[i] = 0;
  for (int i = t; i < NUM_SEG * NCH; i += WGS) red[i] = 0.0f;

  v8f acc0 = {}, acc1 = {}, acc2 = {}, acc3 = {};

  const long base = (long)blockIdx.x * PIX_PER_WG;

  for (int tile = 0; tile < TILES_PER_WG; ++tile) {
    __syncthreads();
    const long p0 = base + (long)tile * TILE;
    {
      const long p = p0 + t;
      const int s = trgt[p] * (mask[p] > 0 ? 1 : 0);
      segl[t] = s;
      atomicAdd(&cnt[s], 1);
    }
#pragma unroll
    for (int c = 0; c < NCH; ++c) {
      // coalesced: lanes read consecutive pixels of one channel stream
      xs[t * LDS_STRIDE + c] = embd[(long)c * NPIX + p0 + t];
      if (tile + 1 < TILES_PER_WG)
        __builtin_prefetch(&embd[(long)c * NPIX + p0 + TILE + t], 0, 1);
    }
    __syncthreads();

    const int lbase = wave * 32;   // each wave consumes 32 pixels of the tile
#pragma unroll
    for (int g = 0; g < 8; ++g) {
      const int l = lbase + g * 4;                  // 4 pixels per WMMA group
      const int4 ss = *(const int4*)&segl[l];
      const int sa0 = hi ? ss.z : ss.x;             // label for my K-slot pair
      const int sa1 = hi ? ss.w : ss.y;
      const int rb0 = hi ? (l + 2) : l;             // B rows per ISA layout
      const int rb1 = hi ? (l + 3) : (l + 1);
      v2f b;
      b.x = xs[rb0 * LDS_STRIDE + m];
      b.y = xs[rb1 * LDS_STRIDE + m];
      v2f a;
      a.x = (sa0 == m) ? 1.0f : 0.0f;
      a.y = (sa1 == m) ? 1.0f : 0.0f;
      acc0 = __builtin_amdgcn_wmma_f32_16x16x4_f32(false, a, false, b,
                                                   (short)0, acc0, false, false);
      a.x = (sa0 == 16 + m) ? 1.0f : 0.0f;
      a.y = (sa1 == 16 + m) ? 1.0f : 0.0f;
      acc1 = __builtin_amdgcn_wmma_f32_16x16x4_f32(false, a, false, b,
                                                   (short)0, acc1, false, false);
      a.x = (sa0 == 32 + m) ? 1.0f : 0.0f;
      a.y = (sa1 == 32 + m) ? 1.0f : 0.0f;
      acc2 = __builtin_amdgcn_wmma_f32_16x16x4_f32(false, a, false, b,
                                                   (short)0, acc2, false, false);
      a.x = (sa0 == 48 + m) ? 1.0f : 0.0f;
      a.y = (sa1 == 48 + m) ? 1.0f : 0.0f;
      acc3 = __builtin_amdgcn_wmma_f32_16x16x4_f32(false, a, false, b,
                                                   (short)0, acc3, false, false);
    }
  }
  __syncthreads();

  // C/D layout: vgpr r, lane -> (M = r + (hi?8:0), N = lane&15)
  const int mrow = hi ? 8 : 0;
#pragma unroll
  for (int r = 0; r < 8; ++r) {
    const int row = r + mrow;
    atomicAdd(&red[(0 * 16 + row) * NCH + m], acc0[r]);
    atomicAdd(&red[(1 * 16 + row) * NCH + m], acc1[r]);
    atomicAdd(&red[(2 * 16 + row) * NCH + m], acc2[r]);
    atomicAdd(&red[(3 * 16 + row) * NCH + m], acc3[r]);
  }
  __syncthreads();
  for (int i = t; i < NUM_SEG * NCH; i += WGS) atomicAdd(&sums[i], red[i]);
  for (int i = t; i < NUM_SEG; i += WGS) atomicAdd(&counts[i], (float)cnt[i]);
}

// -------------------------------------------------- means + present flags
__global__ void means_kernel(const float* __restrict__ sums,
                             const float* __restrict__ counts,
                             float* __restrict__ means,
                             float* __restrict__ present) {
  int i = blockIdx.x * blockDim.x + threadIdx.x;
  if (i < NUM_SEG * NCH) means[i] = sums[i] / fmaxf(counts[i / NCH], 1.0f);
  if (i < NUM_SEG) present[i] = (counts[i] > 0.0f && i != 0) ? 1.0f : 0.0f;
}

// ---------------------------------- pass 2: per-pixel hinge, segment sums
__global__ __launch_bounds__(WGS) void hinge_kernel(
    const float* __restrict__ embd, const int* __restrict__ trgt,
    const int* __restrict__ mask, const float* __restrict__ means,
    float* __restrict__ hinge) {
  __shared__ float mL[NUM_SEG * NCH];
  __shared__ float hl[NUM_SEG];
  const int t = threadIdx.x;
  for (int i = t; i < NUM_SEG * NCH; i += WGS) mL[i] = means[i];
  for (int i = t; i < NUM_SEG; i += WGS) hl[i] = 0.0f;
  __syncthreads();
  const long base = (long)blockIdx.x * PIX_PER_WG;
  for (int tile = 0; tile < TILES_PER_WG; ++tile) {
    const long p = base + (long)tile * TILE + t;
    const int s = trgt[p] * (mask[p] > 0 ? 1 : 0);
    float acc = 0.0f;
#pragma unroll
    for (int c = 0; c < NCH; ++c)
      acc += fabsf(embd[(long)c * NPIX + p] - mL[s * NCH + c]);
    float mg = fmaxf(acc - DELTA_V, 0.0f);
    atomicAdd(&hl[s], mg * mg);
  }
  __syncthreads();
  for (int i = t; i < NUM_SEG; i += WGS) atomicAdd(&hinge[i], hl[i]);
}

// ----------------------------------------------------------- final scalar
__global__ __launch_bounds__(WGS) void final_kernel(
    const float* __restrict__ means, const float* __restrict__ counts,
    const float* __restrict__ hinge, const float* __restrict__ present,
    float* __restrict__ out) {
  __shared__ float mL[NUM_SEG * NCH];
  __shared__ float pr[NUM_SEG];
  __shared__ float rbuf[WGS];
  __shared__ float res[4];
  const int t = threadIdx.x;
  for (int i = t; i < NUM_SEG * NCH; i += WGS) mL[i] = means[i];
  for (int i = t; i < NUM_SEG; i += WGS) pr[i] = present[i];
  __syncthreads();

  float Kp = 0.0f, li = 0.0f, ln = 0.0f, le = 0.0f;
  if (t < NUM_SEG) {
    Kp = pr[t];
    if (pr[t] > 0.0f) {
      li = hinge[t] / fmaxf(counts[t], 1.0f);
      float s = 0.0f;
#pragma unroll
      for (int c = 0; c < NCH; ++c) s += fabsf(mL[t * NCH + c]);
      ln = s;
    }
  }
  for (int k = 0; k < 16; ++k) {  // 256 threads x 16 = 4096 = 64x64 pairs
    const int pid = t * 16 + k;
    const int i = pid >> 6, j = pid & 63;
    if (i != j && pr[i] > 0.0f && pr[j] > 0.0f) {
      float d = 0.0f;
#pragma unroll
      for (int c = 0; c < NCH; ++c) d += fabsf(mL[i * NCH + c] - mL[j * NCH + c]);
      const float mm = fmaxf(2.0f * DELTA_D - d, 0.0f);
      le += mm * mm;
    }
  }
  float vals[4] = {Kp, li, ln, le};
  for (int k = 0; k < 4; ++k) {
    __syncthreads();
    rbuf[t] = vals[k];
    __syncthreads();
    for (int s = WGS / 2; s > 0; s >>= 1) {
      if (t < s) rbuf[t] += rbuf[t + s];
      __syncthreads();
    }
    if (t == 0) res[k] = rbuf[0];
  }
  __syncthreads();
  if (t == 0) {
    const float K = res[0];
    const float dK = fmaxf(1.0f, K);
    const float loss = ALPHA * (res[1] / dK) +
                       BETA * (res[3] / fmaxf(1.0f, K * (K - 1.0f))) +
                       GAMMA * (res[2] / dK);
    out[0] = loss;
  }
}

extern "C" void kernel_launch(void* const* d_in, const int* in_sizes, int n_in,
                              void* d_out, int out_size, void* d_ws,
                              size_t ws_size, hipStream_t stream) {
  const float* embd = (const float*)d_in[0];
  const int* trgt = (const int*)d_in[1];
  const int* mask = (const int*)d_in[2];
  float* ws = (float*)d_ws;
  float* sums = ws;               // 1024
  float* counts = ws + 1024;      // 64
  float* means = ws + 1088;       // 1024
  float* hinge = ws + 2112;       // 64
  float* present = ws + 2176;     // 64 (total 2240 floats)
  float* out = (float*)d_out;

  zero_kernel<<<(2240 + 255) / 256, 256, 0, stream>>>(ws, 2240, out);
  seg_sum_wmma<<<NWG, WGS, 0, stream>>>(embd, trgt, mask, sums, counts);
  means_kernel<<<4, 256, 0, stream>>>(sums, counts, means, present);
  hinge_kernel<<<NWG, WGS, 0, stream>>>(embd, trgt, mask, means, hinge);
  final_kernel<<<1, WGS, 0, stream>>>(means, counts, hinge, present, out);
}